// NeighborlistBruteNsq_17282948399689
// MI455X (gfx1250) — compile-verified
//
#include <hip/hip_runtime.h>
#include <stdint.h>

typedef float v4f __attribute__((ext_vector_type(4)));

#define TILE     128
#define THREADS  256
#define CUTOFF_R 0.5f

// Generic pointers to LDS carry the LDS byte address in their low 32 bits
// (ISA: aperture decode).
__device__ __forceinline__ uint32_t lds_addr32(const void* p) {
  return (uint32_t)(uintptr_t)p;
}

// gfx1250 async copy: global memory -> LDS, 16 bytes per active lane.
// VDST = VGPR holding LDS byte address, VADDR = 64-bit global address, no SADDR.
__device__ __forceinline__ void async_load_b128(uint32_t lds_off, const void* gaddr) {
  asm volatile("global_load_async_to_lds_b128 %0, %1, off"
               :
               : "v"(lds_off), "v"(gaddr)
               : "memory");
}

__device__ __forceinline__ void wait_async_lds() {
  asm volatile("s_wait_asynccnt 0" ::: "memory");
}

// jnp.remainder(x, L) = x - floor(x/L)*L  (result sign of divisor)
__device__ __forceinline__ float pbc_wrap(float r, float L, float invL, float h) {
  float x = r + h;
  x = __builtin_fmaf(-L, floorf(x * invL), x);
  return x - h;
}

__global__ __launch_bounds__(THREADS)
void NeighborlistBruteNsq_kernel(const float* __restrict__ pos,
                                 const float* __restrict__ box,
                                 float* __restrict__ out,
                                 int N, int T) {
  __shared__ __align__(16) float sA[TILE * 3];  // row-atom positions (i side)
  __shared__ __align__(16) float sB[TILE * 3];  // col-atom positions (j side)

  // Map linear block id -> upper-triangle tile (ti, tj), tj >= ti.
  int b  = blockIdx.x;
  int ti = 0;
  while (b >= (T - ti)) { b -= (T - ti); ++ti; }
  int tj = ti + b;
  const int i0 = ti * TILE;
  const int j0 = tj * TILE;
  const int t  = threadIdx.x;

  // Stage the two 128-atom position tiles (1536 B each) into LDS.
  if ((i0 + TILE) <= N && (j0 + TILE) <= N) {
    // 96 chunks of 16 B per side; lanes 0..95 copy A, 128..223 copy B.
    if (t < (TILE * 12 / 16)) {
      async_load_b128(lds_addr32(sA) + (uint32_t)t * 16u,
                      (const char*)pos + (size_t)i0 * 12 + (size_t)t * 16);
    } else if (t >= 128 && t < 128 + (TILE * 12 / 16)) {
      int c = t - 128;
      async_load_b128(lds_addr32(sB) + (uint32_t)c * 16u,
                      (const char*)pos + (size_t)j0 * 12 + (size_t)c * 16);
    }
    wait_async_lds();          // each wave drains its own ASYNCcnt
  } else {
    // ragged tail tile (not hit for N % 128 == 0) — plain staging
    for (int k = t; k < TILE * 3; k += THREADS) {
      int ga = i0 * 3 + k;
      sA[k] = (ga < N * 3) ? pos[ga] : 0.0f;
      int gb = j0 * 3 + k;
      sB[k] = (gb < N * 3) ? pos[gb] : 0.0f;
    }
  }
  __syncthreads();             // cross-wave visibility of LDS tiles

  const float Lx = box[0], Ly = box[4], Lz = box[8];
  const float iLx = 1.0f / Lx, iLy = 1.0f / Ly, iLz = 1.0f / Lz;
  const float hx = 0.5f * Lx, hy = 0.5f * Ly, hz = 0.5f * Lz;

  // Thread owns one column j; its position stays in registers.
  const int lj = t & (TILE - 1);
  const int j  = j0 + lj;
  const float bx = sB[lj * 3 + 0];
  const float by = sB[lj * 3 + 1];
  const float bz = sB[lj * 3 + 2];

  const int liBase = t >> 7;   // 0 or 1: two tile rows per pass
  const long long Nm1 = (long long)(N - 1);

#pragma unroll 4
  for (int pass = 0; pass < TILE / 2; ++pass) {
    const int li = liBase + pass * 2;
    const int i  = i0 + li;

    // Broadcast LDS read (same address across each wave).
    const float ax = sA[li * 3 + 0];
    const float ay = sA[li * 3 + 1];
    const float az = sA[li * 3 + 2];

    float rx = pbc_wrap(ax - bx, Lx, iLx, hx);
    float ry = pbc_wrap(ay - by, Ly, iLy, hy);
    float rz = pbc_wrap(az - bz, Lz, iLz, hz);
    float d  = sqrtf(rx * rx + ry * ry + rz * rz);

    const bool in = (d <= CUTOFF_R);
    v4f val;
    val.x = in ? rx : 0.0f;
    val.y = in ? ry : 0.0f;
    val.z = in ? rz : 0.0f;
    val.w = in ? d  : 0.0f;

    if (j > i && j < N && i < N) {
      // flat triu index: rowstart(i) + (j - i - 1)
      const long long rs = (long long)i * Nm1 - ((long long)i * (i - 1)) / 2;
      const long long p  = rs + (long long)(j - i - 1);
      // Streaming store: output (302 MB) is write-once, keep it out of L2.
      __builtin_nontemporal_store(val, (v4f*)out + p);
    }
  }
}

extern "C" void kernel_launch(void* const* d_in, const int* in_sizes, int n_in,
                              void* d_out, int out_size, void* d_ws, size_t ws_size,
                              hipStream_t stream) {
  const float* pos = (const float*)d_in[0];   // [N,3] f32
  const float* box = (const float*)d_in[1];   // [3,3] f32
  // d_in[2]/d_in[3] are pair_i/pair_j == triu_indices(N,1): deterministic,
  // derived arithmetically on-device to avoid 151 MB of index reads.
  float* out = (float*)d_out;                 // [P,4] f32

  const int N = in_sizes[0] / 3;
  const int T = (N + TILE - 1) / TILE;
  const int nblocks = T * (T + 1) / 2;

  NeighborlistBruteNsq_kernel<<<dim3(nblocks), dim3(THREADS), 0, stream>>>(
      pos, box, out, N, T);
}